// PriorLoss_62749472194941
// MI455X (gfx1250) — compile-verified
//
#include <hip/hip_runtime.h>
#include <stdint.h>

typedef __attribute__((ext_vector_type(2))) float v2f;
typedef __attribute__((ext_vector_type(8))) float v8f;

#define B_  32
#define N_  2048
#define M_  2048
#define NCLS 6
#define CHUNKS 8              // N / 256
#define ROWS_PER_BLOCK 256
#define MTILES (M_ / 16)
#define CD_WT 5.0f

// monotone float <-> uint key (unsigned order == float order)
__device__ __forceinline__ uint32_t fkey(float f) {
    uint32_t u = __float_as_uint(f);
    return (u & 0x80000000u) ? ~u : (u | 0x80000000u);
}
__device__ __forceinline__ float funkey(uint32_t k) {
    uint32_t u = (k & 0x80000000u) ? (k & 0x7FFFFFFFu) : ~k;
    return __uint_as_float(u);
}

// ---------------------------------------------------------------------------
// Kernel 1: per (batch, 256-row chunk): pairwise sq-dists via WMMA f32 16x16x4
//   Trick: A' = (-2x,-2y,-2z,|a|^2), B' = (x,y,z,1), C = broadcast(|b|^2)
//   => WMMA output D[i][j] = |a_i|^2 + |b_j|^2 - 2 a_i.b_j  (exact sq dist)
// ---------------------------------------------------------------------------
__global__ __launch_bounds__(256) void chamfer_tiles(
    const float* __restrict__ inst,      // [B,N,3]
    const float* __restrict__ model,     // [B,M,3]
    uint32_t* __restrict__ colkeys,      // [B,CHUNKS,M]
    float* __restrict__ part1)           // [B*CHUNKS]
{
    __shared__ v2f      s_bfrag[MTILES * 32];   // 32 KB: B' fragments per (tile,lane)
    __shared__ float    s_b2[M_];               //  8 KB
    __shared__ uint32_t s_ck[M_];               //  8 KB
    __shared__ float    s_a2[ROWS_PER_BLOCK];   //  1 KB
    __shared__ float    s_sum1;

    const int tid = threadIdx.x;
    const int b   = blockIdx.x / CHUNKS;
    const int c   = blockIdx.x % CHUNKS;
    const int n0  = c * ROWS_PER_BLOCK;

    if (tid == 0) s_sum1 = 0.0f;
    {   // |a|^2 for this chunk's 256 rows
        const float* p = inst + ((size_t)b * N_ + n0 + tid) * 3;
        float x = p[0], y = p[1], z = p[2];
        s_a2[tid] = x * x + y * y + z * z;
    }
    for (int m = tid; m < M_; m += 256) {   // |b|^2 for all M, init col keys
        const float* p = model + ((size_t)b * M_ + m) * 3;
        float x = p[0], y = p[1], z = p[2];
        s_b2[m] = x * x + y * y + z * z;
        s_ck[m] = 0xFFFFFFFFu;
    }
    // B' fragments: lane half 0 -> (x,y), half 1 -> (z, 1.0)
    for (int idx = tid; idx < MTILES * 32; idx += 256) {
        const int ln  = idx & 31;
        const int mt  = idx >> 5;
        const int gm  = mt * 16 + (ln & 15);
        const float* p = model + ((size_t)b * M_ + gm) * 3;
        v2f f;
        if ((ln >> 4) == 0) { f.x = p[0]; f.y = p[1];  }
        else                { f.x = p[2]; f.y = 1.0f;  }
        s_bfrag[idx] = f;
    }
    __syncthreads();

    const int lane = tid & 31;
    const int wave = tid >> 5;
    const int half = lane >> 4;   // 0: K=0,1 lanes; 1: K=2,3 lanes
    const int l16  = lane & 15;

    // A' fragments (two 16-row tiles per wave): half0 (-2x,-2y), half1 (-2z,|a|^2)
    v2f A[2];
#pragma unroll
    for (int nt = 0; nt < 2; ++nt) {
        const int rowoff = wave * 32 + nt * 16;
        const float* p = inst + ((size_t)b * N_ + n0 + rowoff + l16) * 3;
        if (half == 0) { A[nt].x = -2.0f * p[0]; A[nt].y = -2.0f * p[1];      }
        else           { A[nt].x = -2.0f * p[2]; A[nt].y = s_a2[rowoff + l16]; }
    }

    v8f rmin[2];
#pragma unroll
    for (int nt = 0; nt < 2; ++nt)
#pragma unroll
        for (int r = 0; r < 8; ++r) rmin[nt][r] = 3.4e38f;

    for (int mt = 0; mt < MTILES; ++mt) {
        const int gm = mt * 16 + l16;
        const v2f Bf = s_bfrag[mt * 32 + lane];   // single uniform ds_load_b64
        const float b2j = s_b2[gm];

        v8f cvec;
#pragma unroll
        for (int r = 0; r < 8; ++r) cvec[r] = b2j;

        float cmin = 3.4e38f;
#pragma unroll
        for (int nt = 0; nt < 2; ++nt) {
            // D[i][j] = |a_i|^2 + |b_j|^2 - 2 a_i.b_j  directly
            v8f d = __builtin_amdgcn_wmma_f32_16x16x4_f32(
                false, A[nt], false, Bf, (short)0, cvec, false, false);
#pragma unroll
            for (int r = 0; r < 8; ++r) {
                rmin[nt][r] = fminf(rmin[nt][r], d[r]);
                cmin = fminf(cmin, d[r]);
            }
        }
        // column min over all 16 rows of both tiles: combine lane halves
        cmin = fminf(cmin, __shfl_xor(cmin, 16, 32));
        // branchless: lanes j and j+16 hit the same address with the same value
        atomicMin(&s_ck[gm], fkey(cmin));
    }

    // dist1: finish row mins across the 16 columns (lanes within each half)
    float waveSum = 0.0f;
#pragma unroll
    for (int nt = 0; nt < 2; ++nt) {
        float s = 0.0f;
#pragma unroll
        for (int r = 0; r < 8; ++r) {
            float v = rmin[nt][r];
            v = fminf(v, __shfl_xor(v, 1, 32));
            v = fminf(v, __shfl_xor(v, 2, 32));
            v = fminf(v, __shfl_xor(v, 4, 32));
            v = fminf(v, __shfl_xor(v, 8, 32));
            s += v;                     // uniform within each 16-lane half
        }
        s += __shfl_xor(s, 16, 32);     // rows 0-7 sum + rows 8-15 sum
        waveSum += s;
    }
    if (lane == 0) atomicAdd(&s_sum1, waveSum);
    __syncthreads();

    uint32_t* dst = colkeys + ((size_t)b * CHUNKS + c) * M_;
    for (int m = tid; m < M_; m += 256) dst[m] = s_ck[m];
    if (tid == 0) part1[blockIdx.x] = s_sum1;
}

// ---------------------------------------------------------------------------
// Kernel 2: reduce col-min chunks -> dist2 sum; sum dist1 partials;
//           cross-entropy; write (total, cd, ce)
// ---------------------------------------------------------------------------
__global__ __launch_bounds__(256) void finalize_loss(
    const uint32_t* __restrict__ colkeys,  // [B,CHUNKS,M]
    const float* __restrict__ part1,       // [B*CHUNKS] == 256
    const float* __restrict__ pred,        // [B,NCLS]
    const int* __restrict__ gt,            // [B]
    float* __restrict__ out)               // [3]
{
    __shared__ float red[256];
    __shared__ float res[2];
    const int tid = threadIdx.x;

    // dist2: min over CHUNKS per (b,m), then sum
    float sum2 = 0.0f;
    for (int idx = tid; idx < B_ * M_; idx += 256) {
        const int b = idx / M_;
        const int m = idx - b * M_;
        const uint32_t* base = colkeys + (size_t)b * CHUNKS * M_ + m;
        uint32_t k = 0xFFFFFFFFu;
#pragma unroll
        for (int c = 0; c < CHUNKS; ++c) {
            uint32_t v = base[(size_t)c * M_];
            k = (v < k) ? v : k;
        }
        sum2 += funkey(k);
    }
    red[tid] = sum2; __syncthreads();
    for (int s = 128; s > 0; s >>= 1) {
        if (tid < s) red[tid] += red[tid + s];
        __syncthreads();
    }
    if (tid == 0) res[0] = red[0];
    __syncthreads();

    // dist1: exactly 256 partials
    red[tid] = part1[tid]; __syncthreads();
    for (int s = 128; s > 0; s >>= 1) {
        if (tid < s) red[tid] += red[tid + s];
        __syncthreads();
    }
    if (tid == 0) res[1] = red[0];
    __syncthreads();

    // cross-entropy
    float nll = 0.0f;
    if (tid < B_) {
        const float* l = pred + tid * NCLS;
        float mx = l[0];
#pragma unroll
        for (int k = 1; k < NCLS; ++k) mx = fmaxf(mx, l[k]);
        float se = 0.0f;
#pragma unroll
        for (int k = 0; k < NCLS; ++k) se += __expf(l[k] - mx);
        nll = (__logf(se) + mx) - l[gt[tid]];
    }
    red[tid] = (tid < B_) ? nll : 0.0f; __syncthreads();
    for (int s = 128; s > 0; s >>= 1) {
        if (tid < s) red[tid] += red[tid + s];
        __syncthreads();
    }
    if (tid == 0) {
        float mean2 = res[0] / (float)(B_ * M_);
        float mean1 = res[1] / (float)(B_ * N_);
        float cd    = mean1 + mean2;
        float ce    = red[0] / (float)B_;
        out[0] = CD_WT * cd + ce;
        out[1] = cd;
        out[2] = ce;
    }
}

extern "C" void kernel_launch(void* const* d_in, const int* in_sizes, int n_in,
                              void* d_out, int out_size, void* d_ws, size_t ws_size,
                              hipStream_t stream) {
    (void)in_sizes; (void)n_in; (void)out_size; (void)ws_size;
    const float* inst  = (const float*)d_in[0];   // [B,N,3]
    const float* model = (const float*)d_in[1];   // [B,M,3]
    const float* pred  = (const float*)d_in[2];   // [B,NCLS]
    const int*   gt    = (const int*)d_in[3];     // [B]
    float*       out   = (float*)d_out;           // [3]

    uint32_t* colkeys = (uint32_t*)d_ws;                              // B*CHUNKS*M u32 = 2 MB
    float*    part1   = (float*)((char*)d_ws + (size_t)B_ * CHUNKS * M_ * sizeof(uint32_t));

    chamfer_tiles<<<dim3(B_ * CHUNKS), dim3(256), 0, stream>>>(inst, model, colkeys, part1);
    finalize_loss<<<dim3(1), dim3(256), 0, stream>>>(colkeys, part1, pred, gt, out);
}